// ReconstructionLoss_85083302134236
// MI455X (gfx1250) — compile-verified
//
#include <hip/hip_runtime.h>
#include <hip/hip_bf16.h>
#include <math.h>

#ifndef __has_builtin
#define __has_builtin(x) 0
#endif

#define B_    4
#define SEQ   2048
#define SM1   2047            // S-1 rows per sequence
#define VOC   32000
#define ROWS  (B_ * SM1)      // 8188
#define T1    320             // threads for row kernel: 320*4*25 = 32000 exactly
#define STEPS 25
#define DEPTH 8               // async ring depth (power of 2)
#define NWAVE (T1 / 32)

// Builtin signature (from hipcc diagnostic): v4i AS1* src, v4i AS3* dst, imm, imm
typedef int v4i __attribute__((ext_vector_type(4)));
typedef __attribute__((address_space(1))) v4i* g_v4i_ptr;   // global pointer to int4
typedef __attribute__((address_space(3))) v4i* l_v4i_ptr;   // LDS pointer to int4

// ---- CDNA5 async global->LDS copy (16B per lane), tracked by ASYNCcnt ----
__device__ __forceinline__ void async_copy_b128(const void* gsrc, void* ldst) {
#if __has_builtin(__builtin_amdgcn_global_load_async_to_lds_b128)
  __builtin_amdgcn_global_load_async_to_lds_b128(
      (g_v4i_ptr)(unsigned long long)gsrc,
      (l_v4i_ptr)(unsigned int)(unsigned long long)ldst,
      0, 0);
#else
  // VDST = LDS byte address (low 32 bits of generic LDS pointer), VADDR = 64-bit global addr
  asm volatile("global_load_async_to_lds_b128 %0, %1, off"
               :: "v"((unsigned int)(unsigned long long)ldst), "v"(gsrc)
               : "memory");
#endif
}

#if __has_builtin(__builtin_amdgcn_s_wait_asynccnt)
#define WAIT_ASYNC_IMM(n) __builtin_amdgcn_s_wait_asynccnt(n)
#else
#define WAIT_ASYNC_IMM(n) asm volatile("s_wait_asynccnt %0" :: "i"(n) : "memory")
#endif

__device__ __forceinline__ void wait_async_upto(int n) {
  switch (n) {
    case 0: WAIT_ASYNC_IMM(0); break;
    case 1: WAIT_ASYNC_IMM(1); break;
    case 2: WAIT_ASYNC_IMM(2); break;
    case 3: WAIT_ASYNC_IMM(3); break;
    case 4: WAIT_ASYNC_IMM(4); break;
    case 5: WAIT_ASYNC_IMM(5); break;
    case 6: WAIT_ASYNC_IMM(6); break;
    default: WAIT_ASYNC_IMM(7); break;
  }
}

// Online log-sum-exp update with a 4-wide vector (1.25 exp/elem)
__device__ __forceinline__ void online4(float4 v, float& m, float& s) {
  float mx = fmaxf(fmaxf(v.x, v.y), fmaxf(v.z, v.w));
  float nm = fmaxf(m, mx);
  float e  = __expf(v.x - nm) + __expf(v.y - nm) +
             __expf(v.z - nm) + __expf(v.w - nm);
  s = __fmaf_rn(s, __expf(m - nm), e);
  m = nm;
}

__device__ __forceinline__ void lse_merge(float& m, float& s, float om, float os) {
  float nm = fmaxf(m, om);
  s = s * __expf(m - nm) + os * __expf(om - nm);
  m = nm;
}

// ---------------- Kernel 1: per-row NLL (one block per row) ----------------
__global__ __launch_bounds__(T1) void ce_row_kernel(const float* __restrict__ y_pred,
                                                    const int*   __restrict__ y_true,
                                                    float*       __restrict__ row_nll) {
  __shared__ float4 stage[DEPTH][T1];     // 40 KB async staging ring
  __shared__ float  wm[NWAVE], wsum[NWAVE];

  const int row = blockIdx.x;
  const int b   = row / SM1;
  const int t   = row - b * SM1;
  const int tid = threadIdx.x;

  const float*  rowp  = y_pred + (size_t)(b * SEQ + t) * VOC; // logits[b, t, :]
  const float4* rowp4 = (const float4*)rowp;

  // Prologue: fill the ring (ASYNCcnt -> DEPTH per wave; each lane owns its slot,
  // so no workgroup barrier is ever needed in the hot loop).
#pragma unroll
  for (int p = 0; p < DEPTH; ++p)
    async_copy_b128(rowp4 + (size_t)p * T1 + tid, &stage[p][tid]);

  // Target logit fetched by thread 0 while the DMA ring runs.
  float tgt_logit = 0.f;
  if (tid == 0) {
    int tg = y_true[b * SEQ + t + 1];     // targets = y_true[:, 1:]
    tgt_logit = rowp[tg];
  }

  float m = -INFINITY, s = 0.f;

  for (int st = 0; st < STEPS; ++st) {
    // outstanding async loads for this wave before the one we need completes
    int ahead = (st + DEPTH <= STEPS) ? DEPTH : (STEPS - st);
    wait_async_upto(ahead - 1);                 // in-order completion => exact
    float4 v = stage[st & (DEPTH - 1)][tid];    // ds_load_b128
    online4(v, m, s);
    if (st + DEPTH < STEPS)
      async_copy_b128(rowp4 + (size_t)(st + DEPTH) * T1 + tid,
                      &stage[st & (DEPTH - 1)][tid]);
  }

  // wave32 reduction of (m, s)
#pragma unroll
  for (int off = 16; off > 0; off >>= 1) {
    float om = __shfl_xor(m, off, 32);
    float os = __shfl_xor(s, off, 32);
    lse_merge(m, s, om, os);
  }
  const int wave = tid >> 5;
  if ((tid & 31) == 0) { wm[wave] = m; wsum[wave] = s; }
  __syncthreads();

  if (tid == 0) {
    float M = wm[0], S = wsum[0];
#pragma unroll
    for (int w = 1; w < NWAVE; ++w) lse_merge(M, S, wm[w], wsum[w]);
    // nll = logsumexp(row) - logit[target]
    row_nll[row] = __logf(S) + M - tgt_logit;
  }
}

// -------- Kernel 2: masked per-sequence mean, then mean over batch ---------
__global__ __launch_bounds__(256) void ce_final_kernel(const float* __restrict__ row_nll,
                                                       const int*   __restrict__ y_true,
                                                       const int*   __restrict__ pad_ptr,
                                                       float*       __restrict__ out) {
  __shared__ float ssum[8], scnt[8];
  const int pad  = pad_ptr[0];
  const int tid  = threadIdx.x;
  const int lane = tid & 31;
  const int wave = tid >> 5;

  float acc = 0.f;  // meaningful on thread 0 only
  for (int b = 0; b < B_; ++b) {
    float sum = 0.f, cnt = 0.f;
    for (int t = tid; t < SM1; t += 256) {
      int tg = y_true[b * SEQ + t + 1];
      if (tg != pad) { sum += row_nll[b * SM1 + t]; cnt += 1.f; }
    }
#pragma unroll
    for (int off = 16; off > 0; off >>= 1) {
      sum += __shfl_xor(sum, off, 32);
      cnt += __shfl_xor(cnt, off, 32);
    }
    if (lane == 0) { ssum[wave] = sum; scnt[wave] = cnt; }
    __syncthreads();
    if (tid == 0) {
      float S = 0.f, C = 0.f;
#pragma unroll
      for (int w = 0; w < 8; ++w) { S += ssum[w]; C += scnt[w]; }
      acc += S / C;
    }
    __syncthreads();
  }
  if (tid == 0) out[0] = acc * (1.0f / B_);
}

extern "C" void kernel_launch(void* const* d_in, const int* in_sizes, int n_in,
                              void* d_out, int out_size, void* d_ws, size_t ws_size,
                              hipStream_t stream) {
  const float* y_pred  = (const float*)d_in[0];
  const int*   y_true  = (const int*)d_in[1];
  const int*   pad_ptr = (const int*)d_in[2];
  float*       row_nll = (float*)d_ws;          // 8188 floats of scratch
  (void)in_sizes; (void)n_in; (void)out_size; (void)ws_size;

  ce_row_kernel<<<ROWS, T1, 0, stream>>>(y_pred, y_true, row_nll);
  ce_final_kernel<<<1, 256, 0, stream>>>(row_nll, y_true, pad_ptr, (float*)d_out);
}